// GraphNetwork_20358144983596
// MI455X (gfx1250) — compile-verified
//
#include <hip/hip_runtime.h>
#include <hip/hip_bf16.h>

// ---------------------------------------------------------------------------
// GraphNetwork on MI455X (gfx1250): all GEMMs via V_WMMA_F32_16X16X4_F32
// (full-precision f32 WMMA). Gathers / scatter-mean / residuals fused into
// the GEMM stage & epilogue. A/B tiles staged with GLOBAL_LOAD_ASYNC_TO_LDS
// (ASYNCcnt), wave32, 8 waves/block, 96x128 output strip per block.
// ---------------------------------------------------------------------------

typedef float v2f __attribute__((ext_vector_type(2)));
typedef float v8f __attribute__((ext_vector_type(8)));
typedef int   v4i __attribute__((ext_vector_type(4)));

typedef __attribute__((address_space(1))) v4i* as1_v4i_ptr;   // global
typedef __attribute__((address_space(3))) v4i* as3_v4i_ptr;   // LDS

#define N_NODES 20000
#define N_EDGES 320000
#define HDIM    128

// GEMM tiling
#define KT   64            // K chunk staged in LDS
#define LDA  (KT + 4)      // 68 floats: float4-aligned, bank-spread
#define LDB  (HDIM + 4)    // 132 floats
#define RT   6             // 16-row tiles per wave
#define ROWS (RT * 16)     // 96 rows per block

// A-operand modes
#define AMODE_PLAIN 0
#define AMODE_EDGE  1      // [e | v[dst] | v[src]]       (K = 384)
#define AMODE_NODE  2      // [agg_scaled | v]             (K = 256)
// Epilogue modes (all apply bias + ReLU)
#define EPI_STORE    0
#define EPI_ADD      1     // out[row] += val   (residual)
#define EPI_ADD_SCAT 2     // out[row] += val  AND atomicAdd agg[dst[row]] += val

// ---- CDNA5 async global->LDS copy (ASYNCcnt-tracked, bypasses VGPRs) ----
__device__ __forceinline__ void async_copy_b128(const float* g, float* lds) {
    __builtin_amdgcn_global_load_async_to_lds_b128(
        (as1_v4i_ptr)(v4i*)(float*)g,
        (as3_v4i_ptr)(v4i*)lds,
        0, 0);
}

template <int AMODE, int EPI>
__global__ __launch_bounds__(256) void gemm_wmma_f32(
    const float* __restrict__ A,       // plain A / e / agg depending on AMODE
    const float* __restrict__ V,       // v (AMODE 1,2)
    const int*   __restrict__ dstIdx,  // AMODE 1, EPI 2
    const int*   __restrict__ srcIdx,  // AMODE 1
    const float* __restrict__ W,       // [K,128] row-major
    const float* __restrict__ bias,    // [128]
    float*       __restrict__ out,     // [M,128]
    float*       __restrict__ agg,     // EPI 2 scatter target [N_NODES,128]
    int M, int K)
{
    __shared__ float sA[ROWS * LDA];
    __shared__ float sB[KT * LDB];
    __shared__ int   sDst[ROWS];
    __shared__ int   sSrc[ROWS];

    const int tid  = threadIdx.x;
    const int lane = tid & 31;
    const int wave = tid >> 5;
    const int half = lane >> 4;     // 0: lanes 0-15, 1: lanes 16-31
    const int l15  = lane & 15;
    const int row0 = blockIdx.x * ROWS;
    const int n0   = wave * 16;     // 8 waves cover N = 128

    // ---- stage gather/scatter indices once per block ----
    if (AMODE == AMODE_EDGE || EPI == EPI_ADD_SCAT) {
        if (tid < ROWS) {
            const int grow = row0 + tid;
            const int g    = grow < M ? grow : M - 1;   // clamp tail rows
            sDst[tid] = dstIdx[g];
            if (AMODE == AMODE_EDGE) sSrc[tid] = srcIdx[g];
        }
        __syncthreads();
    }

    v8f acc[RT];
#pragma unroll
    for (int r = 0; r < RT; ++r)
#pragma unroll
        for (int i = 0; i < 8; ++i) acc[r][i] = 0.0f;

    for (int k0 = 0; k0 < K; k0 += KT) {
        const int chunk = (K - k0) < KT ? (K - k0) : KT;
        const int c4n   = chunk >> 2;   // float4s per row of the A chunk

        // ---- async-stage A chunk (gather/concat fused; OOB rows clamped:
        //      garbage rows only feed guarded, never-stored outputs) ----
        for (int t = tid; t < ROWS * c4n; t += 256) {
            const int r    = t / c4n;
            const int c4   = t - r * c4n;
            const int col  = k0 + (c4 << 2);
            const int grow = row0 + r;
            const int gc   = grow < M ? grow : M - 1;
            const float* src;
            if (AMODE == AMODE_PLAIN) {
                src = A + (size_t)gc * K + col;
            } else if (AMODE == AMODE_EDGE) {
                const int seg = col >> 7, w = col & 127;
                if      (seg == 0) src = A + (size_t)gc * HDIM + w;
                else if (seg == 1) src = V + (size_t)sDst[r] * HDIM + w;
                else               src = V + (size_t)sSrc[r] * HDIM + w;
            } else { // AMODE_NODE: [agg_scaled | v]
                const int seg = col >> 7, w = col & 127;
                src = (seg == 0) ? (A + (size_t)gc * HDIM + w)
                                 : (V + (size_t)gc * HDIM + w);
            }
            async_copy_b128(src, &sA[r * LDA + (col - k0)]);
        }
        // ---- async-stage B chunk [chunk x 128] ----
        for (int t = tid; t < (chunk << 5); t += 256) {
            const int kr = t >> 5;
            const int c4 = (t & 31) << 2;
            async_copy_b128(&W[(size_t)(k0 + kr) * HDIM + c4],
                            &sB[kr * LDB + c4]);
        }
        // ---- prefetch next W chunk toward L2 (global_prefetch_b8) ----
        if (k0 + KT < K) {
            __builtin_prefetch(&W[(size_t)(k0 + KT) * HDIM + tid * 32], 0, 3);
        }
        __builtin_amdgcn_s_wait_asynccnt(0);
        __syncthreads();

        // ---- WMMA f32 16x16x4 over the chunk ----
        for (int kk = 0; kk < chunk; kk += 4) {
            // B fragment: vgpr0 = B[kk+2h][n], vgpr1 = B[kk+2h+1][n]
            v2f b;
            b.x = sB[(kk + 2 * half)     * LDB + n0 + l15];
            b.y = sB[(kk + 2 * half + 1) * LDB + n0 + l15];
#pragma unroll
            for (int rt = 0; rt < RT; ++rt) {
                // A fragment: row = l15 (of this 16-row tile), K pair by half
                v2f a = *(const v2f*)&sA[(rt * 16 + l15) * LDA + kk + 2 * half];
                acc[rt] = __builtin_amdgcn_wmma_f32_16x16x4_f32(
                    false, a, false, b, (short)0, acc[rt], false, false);
            }
        }
        __syncthreads();
    }

    // ---- epilogue: bias + ReLU (+ residual / scatter) ----
    const int   col = n0 + l15;
    const float bv  = bias[col];

    if (row0 + ROWS <= M) {
        // fast path: whole block in range, no per-element guards
#pragma unroll
        for (int rt = 0; rt < RT; ++rt) {
            const int rbase = rt * 16 + half * 8;          // C layout: M = i + 8*half
            float* po = out + (size_t)(row0 + rbase) * HDIM + col;
#pragma unroll
            for (int i = 0; i < 8; ++i) {
                float val = fmaxf(acc[rt][i] + bv, 0.0f);
                if (EPI == EPI_STORE) {
                    po[(size_t)i * HDIM] = val;
                } else {
                    po[(size_t)i * HDIM] += val;
                    if (EPI == EPI_ADD_SCAT) {
                        const int d = sDst[rbase + i];
                        atomicAdd(&agg[(size_t)d * HDIM + col], val);
                    }
                }
            }
        }
    } else {
        // tail block: per-row guard
#pragma unroll
        for (int rt = 0; rt < RT; ++rt) {
#pragma unroll
            for (int i = 0; i < 8; ++i) {
                const int rloc = rt * 16 + half * 8 + i;
                const int row  = row0 + rloc;
                if (row < M) {
                    float val = fmaxf(acc[rt][i] + bv, 0.0f);
                    if (EPI == EPI_STORE) {
                        out[(size_t)row * HDIM + col] = val;
                    } else {
                        out[(size_t)row * HDIM + col] += val;
                        if (EPI == EPI_ADD_SCAT) {
                            const int d = sDst[rloc];
                            atomicAdd(&agg[(size_t)d * HDIM + col], val);
                        }
                    }
                }
            }
        }
    }
}

// ---------------------------------------------------------------------------
// Small helper kernels
// ---------------------------------------------------------------------------
__global__ void zero_f32(float* p, int n) {
    int i = blockIdx.x * blockDim.x + threadIdx.x;
    if (i < n) p[i] = 0.0f;
}
__global__ void zero_i32(int* p, int n) {
    int i = blockIdx.x * blockDim.x + threadIdx.x;
    if (i < n) p[i] = 0;
}
__global__ void count_dst(const int* __restrict__ dst, int* __restrict__ cnt, int e) {
    int i = blockIdx.x * blockDim.x + threadIdx.x;
    if (i < e) atomicAdd(&cnt[dst[i]], 1);
}
__global__ void inv_count(const int* __restrict__ cnt, float* __restrict__ inv, int n) {
    int i = blockIdx.x * blockDim.x + threadIdx.x;
    if (i < n) inv[i] = 1.0f / fmaxf((float)cnt[i], 1.0f);
}
// agg[row][*] *= inv[row]  (scatter-sum -> scatter-mean)
__global__ void scale_rows(float* __restrict__ agg, const float* __restrict__ inv, int n) {
    int i = blockIdx.x * blockDim.x + threadIdx.x;
    if (i < n) agg[i] *= inv[i >> 7];
}
// Decoder last layer: [N,128] @ [128,3] + b, no activation (tiny: VALU is fine)
__global__ void dec_final(const float* __restrict__ t, const float* __restrict__ W,
                          const float* __restrict__ b, float* __restrict__ out, int n) {
    int row = blockIdx.x * blockDim.x + threadIdx.x;
    if (row >= n) return;
    float a0 = b[0], a1 = b[1], a2 = b[2];
    const float* x = t + (size_t)row * HDIM;
#pragma unroll 4
    for (int k = 0; k < HDIM; ++k) {
        const float xv = x[k];
        a0 += xv * W[k * 3 + 0];
        a1 += xv * W[k * 3 + 1];
        a2 += xv * W[k * 3 + 2];
    }
    out[row * 3 + 0] = a0;
    out[row * 3 + 1] = a1;
    out[row * 3 + 2] = a2;
}

// ---------------------------------------------------------------------------
// Launch
// ---------------------------------------------------------------------------
static inline int cdiv(int a, int b) { return (a + b - 1) / b; }

extern "C" void kernel_launch(void* const* d_in, const int* in_sizes, int n_in,
                              void* d_out, int out_size, void* d_ws, size_t ws_size,
                              hipStream_t stream) {
    (void)in_sizes; (void)n_in; (void)out_size; (void)ws_size;

    // ---- inputs (setup_inputs dict order, pytrees flattened in order) ----
    const float* node_attr = (const float*)d_in[0];   // [20000,16]
    const float* edge_attr = (const float*)d_in[1];   // [320000,8]
    const int*   eidx      = (const int*)d_in[2];     // [2,320000]
    const int*   srcI = eidx;                         // edge_index[0] = sender
    const int*   dstI = eidx + N_EDGES;               // edge_index[1] = receiver
    const float* encnW1 = (const float*)d_in[3];
    const float* encnB1 = (const float*)d_in[4];
    const float* encnW2 = (const float*)d_in[5];
    const float* encnB2 = (const float*)d_in[6];
    const float* enceW1 = (const float*)d_in[7];
    const float* enceB1 = (const float*)d_in[8];
    const float* enceW2 = (const float*)d_in[9];
    const float* enceB2 = (const float*)d_in[10];
    // layers: per layer 8 arrays (edge W1,b1,W2,b2, node W1,b1,W2,b2) at 11 + l*8
    const float* decW1 = (const float*)d_in[27];
    const float* decB1 = (const float*)d_in[28];
    const float* decW2 = (const float*)d_in[29];
    const float* decB2 = (const float*)d_in[30];

    // ---- workspace carve ----
    float* ws   = (float*)d_ws;
    float* v    = ws;                          // [20000,128]
    float* e    = v   + (size_t)N_NODES * HDIM;
    float* t_e  = e   + (size_t)N_EDGES * HDIM; // edge MLP hidden [320000,128]
    float* agg  = t_e + (size_t)N_EDGES * HDIM; // [20000,128]
    float* t_v  = agg + (size_t)N_NODES * HDIM; // node MLP hidden / dec hidden
    float* inv  = t_v + (size_t)N_NODES * HDIM; // [20000]
    int*   cnt  = (int*)(inv + N_NODES);        // [20000]

    const dim3 blk(256);
    const dim3 gN(cdiv(N_NODES, ROWS));
    const dim3 gE(cdiv(N_EDGES, ROWS));

    // ---- receiver counts -> 1/max(cnt,1) ----
    zero_i32<<<cdiv(N_NODES, 256), blk, 0, stream>>>(cnt, N_NODES);
    count_dst<<<cdiv(N_EDGES, 256), blk, 0, stream>>>(dstI, cnt, N_EDGES);
    inv_count<<<cdiv(N_NODES, 256), blk, 0, stream>>>(cnt, inv, N_NODES);

    // ---- encoders (ReLU after both layers) ----
    gemm_wmma_f32<AMODE_PLAIN, EPI_STORE><<<gN, blk, 0, stream>>>(
        node_attr, nullptr, nullptr, nullptr, encnW1, encnB1, t_v, nullptr, N_NODES, 16);
    gemm_wmma_f32<AMODE_PLAIN, EPI_STORE><<<gN, blk, 0, stream>>>(
        t_v, nullptr, nullptr, nullptr, encnW2, encnB2, v, nullptr, N_NODES, HDIM);
    gemm_wmma_f32<AMODE_PLAIN, EPI_STORE><<<gE, blk, 0, stream>>>(
        edge_attr, nullptr, nullptr, nullptr, enceW1, enceB1, t_e, nullptr, N_EDGES, 8);
    gemm_wmma_f32<AMODE_PLAIN, EPI_STORE><<<gE, blk, 0, stream>>>(
        t_e, nullptr, nullptr, nullptr, enceW2, enceB2, e, nullptr, N_EDGES, HDIM);

    // ---- message-passing layers ----
    for (int l = 0; l < 2; ++l) {
        const float* eW1 = (const float*)d_in[11 + l * 8 + 0];
        const float* eB1 = (const float*)d_in[11 + l * 8 + 1];
        const float* eW2 = (const float*)d_in[11 + l * 8 + 2];
        const float* eB2 = (const float*)d_in[11 + l * 8 + 3];
        const float* nW1 = (const float*)d_in[11 + l * 8 + 4];
        const float* nB1 = (const float*)d_in[11 + l * 8 + 5];
        const float* nW2 = (const float*)d_in[11 + l * 8 + 6];
        const float* nB2 = (const float*)d_in[11 + l * 8 + 7];

        zero_f32<<<cdiv(N_NODES * HDIM, 256), blk, 0, stream>>>(agg, N_NODES * HDIM);

        // edge MLP layer 1: [e | v[dst] | v[src]] (E x 384) @ W -> t_e, ReLU
        gemm_wmma_f32<AMODE_EDGE, EPI_STORE><<<gE, blk, 0, stream>>>(
            e, v, dstI, srcI, eW1, eB1, t_e, nullptr, N_EDGES, 3 * HDIM);
        // edge MLP layer 2: e += relu(...); scatter-add e_new into agg[dst]
        gemm_wmma_f32<AMODE_PLAIN, EPI_ADD_SCAT><<<gE, blk, 0, stream>>>(
            t_e, nullptr, dstI, nullptr, eW2, eB2, e, agg, N_EDGES, HDIM);
        // scatter-sum -> scatter-mean
        scale_rows<<<cdiv(N_NODES * HDIM, 256), blk, 0, stream>>>(agg, inv, N_NODES * HDIM);
        // node MLP layer 1: [agg | v] (N x 256) @ W -> t_v, ReLU
        gemm_wmma_f32<AMODE_NODE, EPI_STORE><<<gN, blk, 0, stream>>>(
            agg, v, nullptr, nullptr, nW1, nB1, t_v, nullptr, N_NODES, 2 * HDIM);
        // node MLP layer 2: v += relu(...)
        gemm_wmma_f32<AMODE_PLAIN, EPI_ADD><<<gN, blk, 0, stream>>>(
            t_v, nullptr, nullptr, nullptr, nW2, nB2, v, nullptr, N_NODES, HDIM);
    }

    // ---- decoder ----
    gemm_wmma_f32<AMODE_PLAIN, EPI_STORE><<<gN, blk, 0, stream>>>(
        v, nullptr, nullptr, nullptr, decW1, decB1, t_v, nullptr, N_NODES, HDIM);
    dec_final<<<cdiv(N_NODES, 256), blk, 0, stream>>>(t_v, decW2, decB2, (float*)d_out, N_NODES);
}